// GATLinkPredNoEmb_56624848830745
// MI455X (gfx1250) — compile-verified
//
#include <hip/hip_runtime.h>
#include <hip/hip_bf16.h>
#include <math.h>

// GATv2 hetero GNN link predictor for MI455X (gfx1250, wave32).
// f32 WMMA 16x16x4 for all dense linear transforms; atomic segment
// max/sum for the edge softmax; everything else is gather/scatter bound.

typedef __attribute__((ext_vector_type(2))) float v2f;
typedef __attribute__((ext_vector_type(8))) float v8f;

#define NEG_SLOPE 0.2f
#define DENOM_EPS 1e-16f

__device__ __forceinline__ v8f wmma_f32_4(v2f a, v2f b, v8f c) {
  // D = A(16x4) x B(4x16) + C(16x16), all f32, wave32
  return __builtin_amdgcn_wmma_f32_16x16x4_f32(
      /*neg_a=*/false, a, /*neg_b=*/false, b,
      /*c_mod=*/(short)0, c, /*reuse_a=*/false, /*reuse_b=*/false);
}

// out[nrows,64] = act(in[nrows,K]) @ W[64,K]^T + bias   (act = optional relu)
// One wave computes a 16x64 tile: 4 accumulators of 16x16, K-loop step 4.
__global__ void __launch_bounds__(128)
gat_gemm_nt_kernel(const float* __restrict__ in, const float* __restrict__ W,
                   const float* __restrict__ bias, float* __restrict__ out,
                   int nrows, int K, int relu_in)
{
  const int lane = threadIdx.x & 31;
  const int wave = threadIdx.x >> 5;
  const int m0 = (blockIdx.x * 4 + wave) * 16;
  if (m0 >= nrows) return;                 // wave-uniform exit; WMMA needs EXEC all-1s
  const int mrow = m0 + (lane & 15);
  const int koff = (lane >> 4) << 1;       // A/B K split: lanes 0-15 -> K+{0,1}, 16-31 -> K+{2,3}
  const int ncol = lane & 15;

  v8f acc[4] = {v8f{}, v8f{}, v8f{}, v8f{}};

  for (int k0 = 0; k0 < K; k0 += 4) {
    float x0 = 0.0f, x1 = 0.0f;
    if (mrow < nrows) {
      const float* ip = in + (size_t)mrow * K + k0 + koff;
      x0 = ip[0];
      x1 = ip[1];
    }
    if (relu_in) { x0 = fmaxf(x0, 0.0f); x1 = fmaxf(x1, 0.0f); }
    v2f a; a.x = x0; a.y = x1;
#pragma unroll
    for (int t = 0; t < 4; ++t) {
      const float* wp = W + (size_t)(t * 16 + ncol) * K + k0 + koff;
      v2f b; b.x = wp[0]; b.y = wp[1];
      acc[t] = wmma_f32_4(a, b, acc[t]);
    }
  }

  // C/D layout: VGPR r holds M=r (lanes 0-15) and M=r+8 (lanes 16-31), N = lane&15
  const int rbase = (lane >> 4) << 3;
#pragma unroll
  for (int r = 0; r < 8; ++r) {
    const int row = m0 + rbase + r;
    if (row < nrows) {
#pragma unroll
      for (int t = 0; t < 4; ++t) {
        out[(size_t)row * 64 + t * 16 + ncol] = acc[t][r] + bias[t * 16 + ncol];
      }
    }
  }
}

// out = bias (per dst node), m = -inf, denom = 0
__global__ void gat_init_kernel(float* __restrict__ outb, float* __restrict__ mbuf,
                                float* __restrict__ dbuf, const float* __restrict__ bias,
                                int n_dst)
{
  int i = blockIdx.x * blockDim.x + threadIdx.x;
  if (i < n_dst * 64) outb[i] = bias[i & 63];
  if (i < n_dst * 4) { mbuf[i] = -__builtin_inff(); dbuf[i] = 0.0f; }
}

__device__ __forceinline__ void atomicMaxF(float* addr, float val) {
  // IEEE-ordered float max via integer atomics (valid with -inf init)
  if (val >= 0.0f) atomicMax((int*)addr, __float_as_int(val));
  else             atomicMin((unsigned int*)addr, __float_as_uint(val));
}

// Pass 1: score[e,h] = att[h] . leaky(xl[src]+xr[dst]); segment max into m
__global__ void gat_edge_score_kernel(const float* __restrict__ xl, const float* __restrict__ xr,
                                      const int* __restrict__ src, const int* __restrict__ dst,
                                      const float* __restrict__ att,
                                      float* __restrict__ sbuf, float* __restrict__ mbuf, int E)
{
  int i = blockIdx.x * blockDim.x + threadIdx.x;
  if (i >= E * 4) return;
  const int e = i >> 2;
  const int h = i & 3;
  const int s = src[e];
  const int d = dst[e];
  const float4* a = (const float4*)(xl + (size_t)s * 64 + h * 16);
  const float4* b = (const float4*)(xr + (size_t)d * 64 + h * 16);
  const float4* w = (const float4*)(att + h * 16);
  float sc = 0.0f;
#pragma unroll
  for (int j = 0; j < 4; ++j) {
    float4 av = a[j], bv = b[j], wv = w[j];
    float v;
    v = av.x + bv.x; v = (v >= 0.0f) ? v : NEG_SLOPE * v; sc += wv.x * v;
    v = av.y + bv.y; v = (v >= 0.0f) ? v : NEG_SLOPE * v; sc += wv.y * v;
    v = av.z + bv.z; v = (v >= 0.0f) ? v : NEG_SLOPE * v; sc += wv.z * v;
    v = av.w + bv.w; v = (v >= 0.0f) ? v : NEG_SLOPE * v; sc += wv.w * v;
  }
  sbuf[i] = sc;
  atomicMaxF(mbuf + (size_t)d * 4 + h, sc);
}

// Pass 2: ex = exp(score - m[dst]); segment sum into denom
__global__ void gat_edge_exp_kernel(float* __restrict__ sbuf, const int* __restrict__ dst,
                                    const float* __restrict__ mbuf, float* __restrict__ dbuf,
                                    int E)
{
  int i = blockIdx.x * blockDim.x + threadIdx.x;
  if (i >= E * 4) return;
  const int e = i >> 2;
  const int h = i & 3;
  const int d = dst[e];
  const float ex = __expf(sbuf[i] - mbuf[(size_t)d * 4 + h]);
  sbuf[i] = ex;
  atomicAdd(dbuf + (size_t)d * 4 + h, ex);
}

// Pass 3: out[dst] += xl[src] * alpha
__global__ void gat_edge_accum_kernel(const float* __restrict__ xl,
                                      const int* __restrict__ src, const int* __restrict__ dst,
                                      const float* __restrict__ sbuf, const float* __restrict__ dbuf,
                                      float* __restrict__ outb, int E)
{
  int i = blockIdx.x * blockDim.x + threadIdx.x;
  if (i >= E * 64) return;
  const int e = i >> 6;
  const int hc = i & 63;
  const int h = hc >> 4;
  const int s = src[e];
  const int d = dst[e];
  const float alpha = sbuf[(size_t)e * 4 + h] / (dbuf[(size_t)d * 4 + h] + DENOM_EPS);
  atomicAdd(outb + (size_t)d * 64 + hc, xl[(size_t)s * 64 + hc] * alpha);
}

// out[i] = dot(xu[label_src[i]], xp[label_dst[i]])
__global__ void gat_dot_kernel(const float* __restrict__ xu, const float* __restrict__ xp,
                               const int* __restrict__ ls, const int* __restrict__ ld,
                               float* __restrict__ out, int n)
{
  int i = blockIdx.x * blockDim.x + threadIdx.x;
  if (i >= n) return;
  const float4* a = (const float4*)(xu + (size_t)ls[i] * 64);
  const float4* b = (const float4*)(xp + (size_t)ld[i] * 64);
  float acc = 0.0f;
#pragma unroll
  for (int j = 0; j < 16; ++j) {
    float4 av = a[j], bv = b[j];
    acc += av.x * bv.x + av.y * bv.y + av.z * bv.z + av.w * bv.w;
  }
  out[i] = acc;
}

extern "C" void kernel_launch(void* const* d_in, const int* in_sizes, int n_in,
                              void* d_out, int out_size, void* d_ws, size_t ws_size,
                              hipStream_t stream) {
  // Input layout (recursive dict-insertion order of setup_inputs):
  // 0:x_user 1:x_prod 2:Wu 3:bu 4:Wp 5:bp
  // 6..41: 3 layers x {buy,rev} x {Wl,bl,Wr,br,att,bias}
  // 42:src_buy 43:dst_buy 44:src_rev 45:dst_rev 46:label_src 47:label_dst
  const float* x_user = (const float*)d_in[0];
  const float* x_prod = (const float*)d_in[1];
  const float* Wu = (const float*)d_in[2];
  const float* bu = (const float*)d_in[3];
  const float* Wp = (const float*)d_in[4];
  const float* bp = (const float*)d_in[5];
  const int* src_buy = (const int*)d_in[42];
  const int* dst_buy = (const int*)d_in[43];
  const int* src_rev = (const int*)d_in[44];
  const int* dst_rev = (const int*)d_in[45];
  const int* label_src = (const int*)d_in[46];
  const int* label_dst = (const int*)d_in[47];

  const int n_user = in_sizes[0] / 32;
  const int n_prod = in_sizes[1] / 64;
  const int E = in_sizes[42];
  const int n_label = in_sizes[46];

  // Workspace layout (floats)
  float* ws = (float*)d_ws;
  size_t off = 0;
  auto wsalloc = [&](size_t n) { float* p = ws + off; off += n; return p; };
  const int n_max = (n_user > n_prod) ? n_user : n_prod;
  float* xu_a = wsalloc((size_t)n_user * 64);
  float* xu_b = wsalloc((size_t)n_user * 64);
  float* xp_a = wsalloc((size_t)n_prod * 64);
  float* xp_b = wsalloc((size_t)n_prod * 64);
  float* xl   = wsalloc((size_t)n_max * 64);
  float* xr   = wsalloc((size_t)n_max * 64);
  float* mb   = wsalloc((size_t)n_max * 4);
  float* db   = wsalloc((size_t)n_max * 4);
  float* sb   = wsalloc((size_t)E * 4);
  (void)ws_size; (void)n_in; (void)out_size;

  auto gemm = [&](const float* in, const float* W, const float* b, float* out,
                  int nrows, int K, int relu) {
    int blocks = (nrows + 63) / 64;    // 4 waves x 16 rows per block
    gat_gemm_nt_kernel<<<blocks, 128, 0, stream>>>(in, W, b, out, nrows, K, relu);
  };

  auto conv = [&](const float* xs, int ns, const float* xd, int nd,
                  const int* src, const int* dst, int pbase, float* outbuf, int relu) {
    const float* Wl   = (const float*)d_in[pbase + 0];
    const float* bl   = (const float*)d_in[pbase + 1];
    const float* Wr   = (const float*)d_in[pbase + 2];
    const float* br   = (const float*)d_in[pbase + 3];
    const float* att  = (const float*)d_in[pbase + 4];
    const float* bias = (const float*)d_in[pbase + 5];

    gemm(xs, Wl, bl, xl, ns, 64, relu);
    gemm(xd, Wr, br, xr, nd, 64, relu);

    int init_threads = nd * 64;
    gat_init_kernel<<<(init_threads + 255) / 256, 256, 0, stream>>>(outbuf, mb, db, bias, nd);

    int e4 = E * 4;
    gat_edge_score_kernel<<<(e4 + 255) / 256, 256, 0, stream>>>(xl, xr, src, dst, att, sb, mb, E);
    gat_edge_exp_kernel<<<(e4 + 255) / 256, 256, 0, stream>>>(sb, dst, mb, db, E);
    int e64 = E * 64;
    gat_edge_accum_kernel<<<(e64 + 255) / 256, 256, 0, stream>>>(xl, src, dst, sb, db, outbuf, E);
  };

  // Input projections
  gemm(x_user, Wu, bu, xu_a, n_user, 32, 0);
  gemm(x_prod, Wp, bp, xp_a, n_prod, 64, 0);

  const float* xu_cur = xu_a; float* xu_nxt = xu_b;
  const float* xp_cur = xp_a; float* xp_nxt = xp_b;
  for (int l = 0; l < 3; ++l) {
    const int relu = (l > 0) ? 1 : 0;  // inter-layer relu applied on consumption
    const int base_buy = 6 + (l * 2 + 0) * 6;
    const int base_rev = 6 + (l * 2 + 1) * 6;
    // new_p = GATv2(xu -> xp over buy edges)
    conv(xu_cur, n_user, xp_cur, n_prod, src_buy, dst_buy, base_buy, xp_nxt, relu);
    // new_u = GATv2(xp -> xu over rev edges)
    conv(xp_cur, n_prod, xu_cur, n_user, src_rev, dst_rev, base_rev, xu_nxt, relu);
    const float* t;
    t = xu_cur; xu_cur = xu_nxt; xu_nxt = (float*)t;
    t = xp_cur; xp_cur = xp_nxt; xp_nxt = (float*)t;
  }

  gat_dot_kernel<<<(n_label + 255) / 256, 256, 0, stream>>>(
      xu_cur, xp_cur, label_src, label_dst, (float*)d_out, n_label);
}